// ORGaNICsCell_14370960573248
// MI455X (gfx1250) — compile-verified
//
#include <hip/hip_runtime.h>

typedef float v2f __attribute__((ext_vector_type(2)));
typedef float v8f __attribute__((ext_vector_type(8)));

#define T_STEPS 1024
#define N_IN    512
#define N_CELLS 4096

// ---------------------------------------------------------------------------
// init: y0 = 0.1, w0 = 1.0, u0 = 0.5
// ---------------------------------------------------------------------------
__global__ void init_state_kernel(float* __restrict__ ybuf0, float* __restrict__ st0) {
  int tid = threadIdx.x;
  for (int j = tid; j < N_CELLS; j += 256) ybuf0[j] = 0.1f;
  if (tid == 0) { st0[0] = 1.0f; st0[1] = 0.5f; }
}

// ---------------------------------------------------------------------------
// Wz [4096,512] row-major  ->  WzT [512,4096] row-major (coalesced writes)
// Wz is only 8 MB so the strided reads stay L2-resident; one-time cost.
// ---------------------------------------------------------------------------
__global__ void transpose_wz_kernel(const float* __restrict__ Wz, float* __restrict__ WzT) {
  int idx = blockIdx.x * 256 + threadIdx.x;  // over 512*4096
  int k = idx >> 12;            // / 4096
  int n = idx & (N_CELLS - 1);  // % 4096
  WzT[idx] = Wz[(size_t)n * N_IN + k];
}

// ---------------------------------------------------------------------------
// Z[1024,4096] = dir_input[1024,512] @ WzT[512,4096] using fp32 WMMA.
// One 16x16 output tile per wave, K-loop of 128 x (K=4) wmma ops.
// A (16x4 f32): lanes 0-15 rows M, VGPR{0,1}=K{0,1}; lanes 16-31: K{2,3}.
// B (4x16 f32): mirrored K-split across lane halves, N = lane&15.
// C/D (16x16 f32): VGPR r holds M = r (lanes 0-15) / M = 8+r (lanes 16-31).
// ---------------------------------------------------------------------------
__global__ __launch_bounds__(256)
void gemm_z_wmma_kernel(const float* __restrict__ X,
                        const float* __restrict__ WzT,
                        float* __restrict__ Z) {
  int wave = threadIdx.x >> 5;
  int lane = threadIdx.x & 31;
  int tile = blockIdx.x * 8 + wave;  // 64 (M) x 256 (N) tiles
  int mt = tile >> 8;
  int nt = tile & 255;
  int m0 = mt << 4, n0 = nt << 4;
  int lm = lane & 15;
  int lh = lane >> 4;

  const float* arow = X   + (size_t)(m0 + lm) * N_IN + (lh << 1);
  const float* bcol = WzT + (size_t)(lh << 1) * N_CELLS + (n0 + lm);

  v8f c = {0.f, 0.f, 0.f, 0.f, 0.f, 0.f, 0.f, 0.f};
  #pragma unroll 4
  for (int k = 0; k < N_IN; k += 4) {
    v2f a, b;
    a.x = arow[k];
    a.y = arow[k + 1];
    b.x = bcol[(size_t)k * N_CELLS];
    b.y = bcol[(size_t)(k + 1) * N_CELLS];
    c = __builtin_amdgcn_wmma_f32_16x16x4_f32(false, a, false, b, (short)0, c,
                                              false, false);
  }

  float* zp = Z + (size_t)(m0 + (lh << 3)) * N_CELLS + n0 + lm;
  #pragma unroll
  for (int r = 0; r < 8; ++r) zp[(size_t)r * N_CELLS] = c[r];
}

// ---------------------------------------------------------------------------
// One launch per timestep. Blocks 0..511: wave-per-row fused Wy/Wx matvecs
// (y+ staged in LDS, float4 b128 loads, shuffle reduction) + y update.
// Block 512: sum(relu(y)^2) reduction + scalar (u, w) pool dynamics.
// Both halves depend ONLY on state t-1, so they run concurrently.
// ---------------------------------------------------------------------------
__global__ __launch_bounds__(256)
void step_kernel(const float* __restrict__ Wy, const float* __restrict__ Wx,
                 const float* __restrict__ Z,
                 const float* __restrict__ a_shift, const float* __restrict__ b_arr,
                 const float* __restrict__ y_old, float* __restrict__ y_new,
                 const float* __restrict__ st_old, float* __restrict__ st_new,
                 float* __restrict__ out, float* __restrict__ us_out,
                 float* __restrict__ ws_out, int t) {
  __shared__ float yp[N_CELLS];
  __shared__ float red[256];
  int tid = threadIdx.x;

  if (blockIdx.x == N_CELLS / 8) {  // scalar pool block
    float s = 0.f;
    for (int i = tid; i < N_CELLS; i += 256) {
      float v = fmaxf(y_old[i], 0.f);
      s += v * v;
    }
    red[tid] = s;
    __syncthreads();
    for (int off = 128; off > 0; off >>= 1) {
      if (tid < off) red[tid] += red[tid + off];
      __syncthreads();
    }
    if (tid == 0) {
      float S = red[0];
      float w_old = st_old[0], u_old = st_old[1];
      float b_plus = fmaxf(b_arr[t], 0.f);
      float w_plus = fmaxf(w_old, 0.f);
      float u_plus = fminf(u_old, 1.f);
      float r = b_plus * 0.1f / (b_plus + u_plus);        // b+ * SIGMA / (b+ + u+)
      float du = 0.1f * (-u_old + u_old * (S + r * r));   // DT/TAU_U = 0.1
      float u_n = u_old + du;
      float du_plus = fmaxf(u_n, 0.f) - u_plus;
      float dw = 0.5f * (-w_old + w_plus * u_plus + 1.0f + du_plus);  // DT/TAU_W = 0.5
      float w_n = w_old + dw;
      st_new[0] = w_n;
      st_new[1] = u_n;
      us_out[t] = u_n;
      ws_out[t] = w_n;
    }
    return;
  }

  // stage relu(y_old) into LDS with coalesced float4 loads
  const float4* y4 = (const float4*)y_old;
  float4* yps = (float4*)yp;
  for (int i = tid; i < N_CELLS / 4; i += 256) {
    float4 v = y4[i];
    v.x = fmaxf(v.x, 0.f); v.y = fmaxf(v.y, 0.f);
    v.z = fmaxf(v.z, 0.f); v.w = fmaxf(v.w, 0.f);
    yps[i] = v;
  }
  __syncthreads();

  int wave = tid >> 5, lane = tid & 31;
  int row = blockIdx.x * 8 + wave;
  const float4* wy4 = (const float4*)(Wy + (size_t)row * N_CELLS);
  const float4* wx4 = (const float4*)(Wx + (size_t)row * N_CELLS);
  const float4* yp4 = (const float4*)yp;

  float accY = 0.f, accX = 0.f;
  #pragma unroll 4
  for (int i = lane; i < N_CELLS / 4; i += 32) {
    float4 wv = wy4[i];
    float4 xv = wx4[i];
    float4 yv = yp4[i];
    accY += wv.x * yv.x + wv.y * yv.y + wv.z * yv.z + wv.w * yv.w;
    accX += xv.x * yv.x + xv.y * yv.y + xv.z * yv.z + xv.w * yv.w;
  }
  // wave32 reduction
  for (int m = 16; m > 0; m >>= 1) {
    accY += __shfl_xor(accY, m, 32);
    accX += __shfl_xor(accX, m, 32);
  }

  if (lane == 0) {
    float yo = y_old[row];
    float b_plus = fmaxf(b_arr[t], 0.f);
    float at = a_shift[t];
    float w_plus = fmaxf(st_old[0], 0.f);     // w+ from OLD w (matches reference)
    float z = Z[(size_t)t * N_CELLS + row];
    float inv1b = 1.0f / (1.0f + b_plus);
    float dy = 0.5f * (-yo + b_plus * inv1b * z + (inv1b / w_plus) * accY) + at * accX;
    float yn = yo + dy;
    y_new[row] = yn;
    out[(size_t)t * N_CELLS + row] = fmaxf(yn, 0.f);
  }
}

// ---------------------------------------------------------------------------
// Host launcher
// ---------------------------------------------------------------------------
extern "C" void kernel_launch(void* const* d_in, const int* in_sizes, int n_in,
                              void* d_out, int out_size, void* d_ws, size_t ws_size,
                              hipStream_t stream) {
  const float* dir_input = (const float*)d_in[0];  // [1024, 512]
  const float* a_shift   = (const float*)d_in[1];  // [1024]
  const float* b_arr     = (const float*)d_in[2];  // [1024]
  const float* Wz        = (const float*)d_in[3];  // [4096, 512]
  const float* Wy        = (const float*)d_in[4];  // [4096, 4096]
  const float* Wx        = (const float*)d_in[5];  // [4096, 4096]

  float* out = (float*)d_out;                       // [1024, 4096]
  float* us  = out + (size_t)T_STEPS * N_CELLS;     // [1024]
  float* wso = us + T_STEPS;                        // [1024]

  char* wsb = (char*)d_ws;
  float* Z   = (float*)wsb;                                   // 16 MB
  float* WzT = (float*)(wsb + (size_t)16 * 1024 * 1024);      // 8 MB
  float* yb0 = (float*)(wsb + (size_t)24 * 1024 * 1024);      // 16 KB
  float* yb1 = yb0 + N_CELLS;                                 // 16 KB
  float* st0 = yb1 + N_CELLS;                                 // 2 floats (+pad)
  float* st1 = st0 + 4;
  float* ybuf[2] = {yb0, yb1};
  float* st[2]   = {st0, st1};

  init_state_kernel<<<1, 256, 0, stream>>>(ybuf[0], st[0]);
  transpose_wz_kernel<<<(N_IN * N_CELLS) / 256, 256, 0, stream>>>(Wz, WzT);
  gemm_z_wmma_kernel<<<(64 * 256) / 8, 256, 0, stream>>>(dir_input, WzT, Z);

  for (int t = 0; t < T_STEPS; ++t) {
    int o = t & 1, nx = o ^ 1;
    step_kernel<<<N_CELLS / 8 + 1, 256, 0, stream>>>(
        Wy, Wx, Z, a_shift, b_arr, ybuf[o], ybuf[nx], st[o], st[nx],
        out, us, wso, t);
  }
}